// ODE4_8392366096838
// MI455X (gfx1250) — compile-verified
//
#include <hip/hip_runtime.h>
#include <hip/hip_bf16.h>
#include <math.h>

// Problem dims (match reference)
#define B_TOT 1024
#define T_LEN 4096

typedef __attribute__((ext_vector_type(2))) float v2f;
typedef __attribute__((ext_vector_type(8))) float v8f;

// CDNA5 f32 WMMA: D(16x16,f32) = A(16x4,f32) x B(4x16,f32) + C
static __device__ __forceinline__ v8f wmma_k4(v2f a, v2f b, v8f c) {
  return __builtin_amdgcn_wmma_f32_16x16x4_f32(
      /*neg_a=*/false, a, /*neg_b=*/false, b,
      /*c_mod=*/(short)0, c, /*reuse_a=*/false, /*reuse_b=*/false);
}

#if defined(__has_builtin)
# if __has_builtin(__builtin_amdgcn_tanhf)
#  define MYTANH(x) __builtin_amdgcn_tanhf(x)
# elif __has_builtin(__builtin_amdgcn_tanh_f32)
#  define MYTANH(x) __builtin_amdgcn_tanh_f32(x)
# else
#  define MYTANH(x) tanhf(x)
# endif
#else
# define MYTANH(x) tanhf(x)
#endif

// One wave (32 lanes) integrates 16 batch rows for all T steps.
// y carried in registers as a WMMA C-tile; LDS used only for the
// C-layout -> A-layout lane transposes between chained matmuls.
__global__ __launch_bounds__(32)
void ode_scan_kernel(const float* __restrict__ x,    // [B,T,8]
                     const float* __restrict__ tarr, // [T]
                     const float* __restrict__ y0,   // [B,8]
                     const float* __restrict__ Wr1,  // [16,32]
                     const float* __restrict__ br1,  // [32]
                     const float* __restrict__ Wr2,  // [32,8]
                     const float* __restrict__ br2,  // [8]
                     const float* __restrict__ W1,   // [8,10]
                     const float* __restrict__ b1,   // [10]
                     const float* __restrict__ W2,   // [10,2]
                     const float* __restrict__ b2,   // [2]
                     float* __restrict__ out)        // [B,T,2]
{
  __shared__ float sY[16 * 8];    // y tile, row-major [row][state]
  __shared__ float sH[16 * 32];   // hidden tile   [row][hidden]
  __shared__ float sP[16 * 16];   // head hidden   [row][unit]

  const int lane = threadIdx.x & 31;
  const int col  = lane & 15;     // N index / A-row index
  const int hi   = lane >> 4;     // lane half: selects K {0,1} vs {2,3}
  const int b0   = blockIdx.x * 16;
  const int arow = col;

  // ---------------- preload weights as B-fragments ----------------
  // B-frag layout (f32, 4x16): v0 = W[k0+2*hi][n], v1 = W[k0+2*hi+1][n]
  v2f bWr1[2][4];                 // [N-tile][K-frag]
  #pragma unroll
  for (int tn = 0; tn < 2; ++tn)
    #pragma unroll
    for (int f = 0; f < 4; ++f) {
      const int k = 4 * f + 2 * hi;
      const int n = col + 16 * tn;
      bWr1[tn][f] = v2f{Wr1[k * 32 + n], Wr1[(k + 1) * 32 + n]};
    }
  v2f bWr2[8];                    // Wr2 [32,8] padded to N=16 with zeros
  #pragma unroll
  for (int f = 0; f < 8; ++f) {
    const int k = 4 * f + 2 * hi;
    const float u0 = (col < 8) ? Wr2[k * 8 + col] : 0.0f;
    const float u1 = (col < 8) ? Wr2[(k + 1) * 8 + col] : 0.0f;
    bWr2[f] = v2f{u0, u1};
  }
  v2f bW1[2];                     // W1 [8,10] padded to N=16 with zeros
  #pragma unroll
  for (int f = 0; f < 2; ++f) {
    const int k = 4 * f + 2 * hi;
    const float u0 = (col < 10) ? W1[k * 10 + col] : 0.0f;
    const float u1 = (col < 10) ? W1[(k + 1) * 10 + col] : 0.0f;
    bW1[f] = v2f{u0, u1};
  }
  float w2c0[10], w2c1[10];
  #pragma unroll
  for (int j = 0; j < 10; ++j) { w2c0[j] = W2[j * 2 + 0]; w2c1[j] = W2[j * 2 + 1]; }
  const float b2r0 = b2[0], b2r1 = b2[1];
  const float br1s0 = br1[col];
  const float br1s1 = br1[col + 16];
  const float br2s  = (col < 8)  ? br2[col] : 0.0f;
  const float b1s   = (col < 10) ? b1[col]  : 0.0f;

  // ---------------- init y as a C-tile ----------------
  // C layout: lane holds column N=col, VGPR v holds row (v + 8*hi)
  v8f y;
  #pragma unroll
  for (int v = 0; v < 8; ++v) {
    const int r = v + 8 * hi;
    y[v] = (col < 8) ? y0[(size_t)(b0 + r) * 8 + col] : 0.0f;
  }

  const float* xrow = x + (size_t)(b0 + arow) * T_LEN * 8;  // this lane's A-row stream

  for (int t = 0; t < T_LEN; ++t) {
    // ---- y C-tile -> LDS -> y A-frags (also the head's A input) ----
    if (col < 8) {
      #pragma unroll
      for (int v = 0; v < 8; ++v) sY[(v + 8 * hi) * 8 + col] = y[v];
    }
    __syncthreads();
    const v2f a0 = *(const v2f*)&sY[arow * 8 + 0 + 2 * hi];  // states 0..3
    const v2f a1 = *(const v2f*)&sY[arow * 8 + 4 + 2 * hi];  // states 4..7
    __syncthreads();

    // ---------------- head: out[:,t,:] = relu(y@W1+b1)@W2+b2 ----------------
    v8f p = v8f{b1s, b1s, b1s, b1s, b1s, b1s, b1s, b1s};
    p = wmma_k4(a0, bW1[0], p);
    p = wmma_k4(a1, bW1[1], p);
    #pragma unroll
    for (int v = 0; v < 8; ++v) p[v] = fmaxf(p[v], 0.0f);
    #pragma unroll
    for (int v = 0; v < 8; ++v) sP[(v + 8 * hi) * 16 + col] = p[v];
    __syncthreads();
    if (lane < 16) {
      const float* pr = &sP[lane * 16];
      float o0 = b2r0, o1 = b2r1;
      #pragma unroll
      for (int j = 0; j < 10; ++j) {
        const float pj = pr[j];
        o0 = fmaf(pj, w2c0[j], o0);
        o1 = fmaf(pj, w2c1[j], o1);
      }
      float* op = out + ((size_t)(b0 + lane) * T_LEN + t) * 2;
      *(v2f*)op = v2f{o0, o1};
    }
    __syncthreads();

    if (t == T_LEN - 1) break;

    // ---- exogenous features (A-frags for K=8..15) straight from global ----
    const float* ep = xrow + (size_t)t * 8;
    const v2f a2 = *(const v2f*)(ep + 0 + 2 * hi);
    const v2f a3 = *(const v2f*)(ep + 4 + 2 * hi);
    __builtin_prefetch(ep + 8 * 8, 0, 0);   // global_prefetch_b8, 8 steps ahead
    const float dt = tarr[t + 1] - tarr[t];

    // ---- matmul1: h = tanh([y,e] @ Wr1 + br1) : two 16x16 N-tiles, K=16 ----
    v8f h0 = v8f{br1s0, br1s0, br1s0, br1s0, br1s0, br1s0, br1s0, br1s0};
    h0 = wmma_k4(a0, bWr1[0][0], h0);
    h0 = wmma_k4(a1, bWr1[0][1], h0);
    h0 = wmma_k4(a2, bWr1[0][2], h0);
    h0 = wmma_k4(a3, bWr1[0][3], h0);
    v8f h1 = v8f{br1s1, br1s1, br1s1, br1s1, br1s1, br1s1, br1s1, br1s1};
    h1 = wmma_k4(a0, bWr1[1][0], h1);
    h1 = wmma_k4(a1, bWr1[1][1], h1);
    h1 = wmma_k4(a2, bWr1[1][2], h1);
    h1 = wmma_k4(a3, bWr1[1][3], h1);
    #pragma unroll
    for (int v = 0; v < 8; ++v) { h0[v] = MYTANH(h0[v]); h1[v] = MYTANH(h1[v]); }

    // ---- h C-tiles -> LDS -> A-frags for matmul2 (K=32) ----
    #pragma unroll
    for (int v = 0; v < 8; ++v) {
      sH[(v + 8 * hi) * 32 + col]      = h0[v];
      sH[(v + 8 * hi) * 32 + col + 16] = h1[v];
    }
    __syncthreads();
    v8f r = v8f{br2s, br2s, br2s, br2s, br2s, br2s, br2s, br2s};
    #pragma unroll
    for (int f = 0; f < 8; ++f) {
      const v2f af = *(const v2f*)&sH[arow * 32 + 4 * f + 2 * hi];
      r = wmma_k4(af, bWr2[f], r);
    }
    __syncthreads();

    // ---- explicit Euler in C-layout (pad columns stay exactly 0) ----
    #pragma unroll
    for (int v = 0; v < 8; ++v) y[v] = fmaf(dt, r[v], y[v]);
  }
}

extern "C" void kernel_launch(void* const* d_in, const int* in_sizes, int n_in,
                              void* d_out, int out_size, void* d_ws, size_t ws_size,
                              hipStream_t stream) {
  const float* x   = (const float*)d_in[0];
  const float* t   = (const float*)d_in[1];
  const float* y0  = (const float*)d_in[2];
  const float* Wr1 = (const float*)d_in[3];
  const float* br1 = (const float*)d_in[4];
  const float* Wr2 = (const float*)d_in[5];
  const float* br2 = (const float*)d_in[6];
  const float* W1  = (const float*)d_in[7];
  const float* b1  = (const float*)d_in[8];
  const float* W2  = (const float*)d_in[9];
  const float* b2  = (const float*)d_in[10];
  float* out = (float*)d_out;
  (void)in_sizes; (void)n_in; (void)out_size; (void)d_ws; (void)ws_size;

  dim3 grid(B_TOT / 16);
  dim3 block(32);
  hipLaunchKernelGGL(ode_scan_kernel, grid, block, 0, stream,
                     x, t, y0, Wr1, br1, Wr2, br2, W1, b1, W2, b2, out);
}